// GRN_conditional_network_77223511982361
// MI455X (gfx1250) — compile-verified
//
#include <hip/hip_runtime.h>
#include <math.h>

#define GENE_NUM 5000
#define GENE_DIM 64
#define HIDDEN 128
#define BATCH 4
#define CT_DIM 16
#define NROWS (BATCH * GENE_NUM)   // 20000
#define EPS 1e-5f

typedef float v2f __attribute__((ext_vector_type(2)));
typedef float v8f __attribute__((ext_vector_type(8)));

__device__ __forceinline__ float silu_f(float v) { return v / (1.f + __expf(-v)); }

__device__ __forceinline__ float wave_sum(float v) {
#pragma unroll
  for (int m = 16; m > 0; m >>= 1) v += __shfl_xor(v, m, 32);
  return v;
}

// monotone float -> uint key so atomicMax(uint) == float max; all real keys > 0
__device__ __forceinline__ unsigned fkey(float f) {
  int i = __float_as_int(f);
  return (i >= 0) ? ((unsigned)i | 0x80000000u) : ~(unsigned)i;
}
__device__ __forceinline__ float funkey(unsigned u) {
  int i = (u & 0x80000000u) ? (int)(u & 0x7fffffffu) : (int)~u;
  return __int_as_float(i);
}

// ---------------------------------------------------------------------------
// WMMA f32 GEMM: Y[M,N] = act(X[M,K] @ W[K,N] + bias). One wave per 16x64 tile.
// grid = (M/16, N/64), block = 32. Uses V_WMMA_F32_16X16X4_F32 (full f32).
// act: 0 = none, 1 = silu
// ---------------------------------------------------------------------------
__global__ void wmma_gemm_f32(const float* __restrict__ X, const float* __restrict__ W,
                              const float* __restrict__ bias, float* __restrict__ Y,
                              int K, int N, int act) {
  const int lane = threadIdx.x;
  const int half = lane >> 4;      // 0: lanes 0-15, 1: lanes 16-31
  const int l15  = lane & 15;
  const int mt   = blockIdx.x;
  const int colBase = blockIdx.y * 64;
  const int row  = mt * 16 + l15;
  const int koff = half * 2;       // K sub-offset per ISA A/B layout

  v8f acc[4];
#pragma unroll
  for (int c = 0; c < 4; ++c)
#pragma unroll
    for (int r = 0; r < 8; ++r) acc[c][r] = 0.f;

  for (int k = 0; k < K; k += 4) {
    // A tile 16x4: lane holds X[row][k+koff], X[row][k+koff+1]
    const float* xp = X + (size_t)row * K + (k + koff);
    v2f a;
    a.x = xp[0];
    a.y = xp[1];
#pragma unroll
    for (int c = 0; c < 4; ++c) {
      int col = colBase + c * 16 + l15;
      // B tile 4x16: lane holds W[k+koff][col], W[k+koff+1][col]
      v2f b;
      b.x = W[(size_t)(k + koff) * N + col];
      b.y = W[(size_t)(k + koff + 1) * N + col];
      acc[c] = __builtin_amdgcn_wmma_f32_16x16x4_f32(false, a, false, b,
                                                     (short)0, acc[c], false, false);
    }
  }

#pragma unroll
  for (int c = 0; c < 4; ++c) {
    int col = colBase + c * 16 + l15;
    float bv = bias ? bias[col] : 0.f;
#pragma unroll
    for (int r = 0; r < 8; ++r) {
      int m = mt * 16 + r + half * 8;   // C/D layout: VGPR r -> rows r / r+8
      float v = acc[c][r] + bv;
      if (act == 1) v = silu_f(v);
      Y[(size_t)m * N + col] = v;
    }
  }
}

// ---------------------------------------------------------------------------
// cell-type encoder: [4,16] -> silu -> [4,64] -> [4,64] -> LayerNorm. 1 block.
// ---------------------------------------------------------------------------
__global__ void ct_mlp_ln_kernel(const float* __restrict__ ct_emb,
                                 const float* __restrict__ w0, const float* __restrict__ b0,
                                 const float* __restrict__ w1, const float* __restrict__ b1,
                                 const float* __restrict__ ln_g, const float* __restrict__ ln_b,
                                 float* __restrict__ ct_ln) {
  __shared__ float sh[64];
  __shared__ float stats[2];
  int d = threadIdx.x;  // 0..63
  for (int b = 0; b < BATCH; ++b) {
    float acc = b0[d];
    for (int i = 0; i < CT_DIM; ++i) acc += ct_emb[b * CT_DIM + i] * w0[i * GENE_DIM + d];
    acc = silu_f(acc);
    __syncthreads();
    sh[d] = acc;
    __syncthreads();
    float h2 = b1[d];
    for (int j = 0; j < 64; ++j) h2 += sh[j] * w1[j * 64 + d];
    __syncthreads();
    sh[d] = h2;
    __syncthreads();
    if (d == 0) {
      float s = 0.f;
      for (int j = 0; j < 64; ++j) s += sh[j];
      float m = s / 64.f, v = 0.f;
      for (int j = 0; j < 64; ++j) { float t = sh[j] - m; v += t * t; }
      stats[0] = m; stats[1] = v / 64.f;
    }
    __syncthreads();
    ct_ln[b * 64 + d] = (h2 - stats[0]) * rsqrtf(stats[1] + EPS) * ln_g[d] + ln_b[d];
  }
}

// x_noisy layer0: out[row,d] = silu(x[row]*w0[d] + b0[d])
__global__ void xn0_kernel(const float* __restrict__ xn, const float* __restrict__ w0,
                           const float* __restrict__ b0, float* __restrict__ out) {
  int idx = blockIdx.x * blockDim.x + threadIdx.x;
  if (idx >= NROWS * 64) return;
  int row = idx >> 6, d = idx & 63;
  out[idx] = silu_f(xn[row] * w0[d] + b0[d]);
}

// LayerNorm over width-64 rows, one wave per row
__global__ void ln_rows64_kernel(const float* __restrict__ in, const float* __restrict__ g,
                                 const float* __restrict__ b, float* __restrict__ out) {
  int row = blockIdx.x, lane = threadIdx.x;
  float x0 = in[(size_t)row * 64 + lane], x1 = in[(size_t)row * 64 + 32 + lane];
  float mean = wave_sum(x0 + x1) / 64.f;
  float d0 = x0 - mean, d1 = x1 - mean;
  float var = wave_sum(d0 * d0 + d1 * d1) / 64.f;
  float rs = rsqrtf(var + EPS);
  out[(size_t)row * 64 + lane]      = d0 * rs * g[lane] + b[lane];
  out[(size_t)row * 64 + 32 + lane] = d1 * rs * g[32 + lane] + b[32 + lane];
}

// gi = LN(xn_row) + ln_ge[n] + ct_ln[b]
__global__ void combine_gi_kernel(const float* __restrict__ xn, const float* __restrict__ lnge,
                                  const float* __restrict__ ctln, const float* __restrict__ g,
                                  const float* __restrict__ b, float* __restrict__ gi) {
  int row = blockIdx.x, lane = threadIdx.x;
  int n = row % GENE_NUM, bb = row / GENE_NUM;
  float x0 = xn[(size_t)row * 64 + lane], x1 = xn[(size_t)row * 64 + 32 + lane];
  float mean = wave_sum(x0 + x1) / 64.f;
  float d0 = x0 - mean, d1 = x1 - mean;
  float var = wave_sum(d0 * d0 + d1 * d1) / 64.f;
  float rs = rsqrtf(var + EPS);
  gi[(size_t)row * 64 + lane] =
      d0 * rs * g[lane] + b[lane] + lnge[(size_t)n * 64 + lane] + ctln[bb * 64 + lane];
  gi[(size_t)row * 64 + 32 + lane] =
      d1 * rs * g[32 + lane] + b[32 + lane] + lnge[(size_t)n * 64 + 32 + lane] + ctln[bb * 64 + 32 + lane];
}

// per-node attention halves: es/ed[row,h] = dot(xw[row, h*64:...], a_{src,dst}[h])
__global__ void attn_coef_kernel(const float* __restrict__ xw, const float* __restrict__ a_src,
                                 const float* __restrict__ a_dst, float* __restrict__ es,
                                 float* __restrict__ ed) {
  int t = blockIdx.x * blockDim.x + threadIdx.x;
  if (t >= NROWS * 2) return;
  int row = t >> 1, h = t & 1;
  const float* xp = xw + (size_t)row * HIDDEN + h * 64;
  const float* as = a_src + h * 64;
  const float* ad = a_dst + h * 64;
  float s = 0.f, d = 0.f;
  for (int c = 0; c < 64; ++c) { float x = xp[c]; s += x * as[c]; d += x * ad[c]; }
  es[t] = s; ed[t] = d;
}

__global__ void zero_kernel(float* __restrict__ p, int n) {
  int i = blockIdx.x * blockDim.x + threadIdx.x;
  if (i < n) p[i] = 0.f;
}

__device__ __forceinline__ void edge_nodes(const int* ei, int n_edges, int e, int& s, int& d) {
  if (e < n_edges) { s = ei[e]; d = ei[n_edges + e]; } else { s = d = e - n_edges; }
}

__global__ void edge_max_kernel(const int* __restrict__ ei, int n_edges,
                                const float* __restrict__ es, const float* __restrict__ ed,
                                unsigned* __restrict__ maxk) {
  int etot = n_edges + GENE_NUM;
  int t = blockIdx.x * blockDim.x + threadIdx.x;
  if (t >= BATCH * etot) return;
  int b = t / etot, e = t % etot, s, d;
  edge_nodes(ei, n_edges, e, s, d);
  int rs = b * GENE_NUM + s, rd = b * GENE_NUM + d;
#pragma unroll
  for (int h = 0; h < 2; ++h) {
    float v = es[rs * 2 + h] + ed[rd * 2 + h];
    v = v > 0.f ? v : 0.2f * v;
    atomicMax(&maxk[rd * 2 + h], fkey(v));
  }
}

__global__ void edge_denom_kernel(const int* __restrict__ ei, int n_edges,
                                  const float* __restrict__ es, const float* __restrict__ ed,
                                  const unsigned* __restrict__ maxk, float* __restrict__ denom) {
  int etot = n_edges + GENE_NUM;
  int t = blockIdx.x * blockDim.x + threadIdx.x;
  if (t >= BATCH * etot) return;
  int b = t / etot, e = t % etot, s, d;
  edge_nodes(ei, n_edges, e, s, d);
  int rs = b * GENE_NUM + s, rd = b * GENE_NUM + d;
#pragma unroll
  for (int h = 0; h < 2; ++h) {
    float v = es[rs * 2 + h] + ed[rd * 2 + h];
    v = v > 0.f ? v : 0.2f * v;
    atomicAdd(&denom[rd * 2 + h], __expf(v - funkey(maxk[rd * 2 + h])));
  }
}

// one block (128 threads) per edge: out[dst, :] += alpha_h * xw[src, :]
__global__ void edge_scatter_kernel(const int* __restrict__ ei, int n_edges,
                                    const float* __restrict__ es, const float* __restrict__ ed,
                                    const unsigned* __restrict__ maxk,
                                    const float* __restrict__ denom,
                                    const float* __restrict__ xw, float* __restrict__ out) {
  int etot = n_edges + GENE_NUM;
  int t = blockIdx.x;
  int b = t / etot, e = t % etot, s, d;
  edge_nodes(ei, n_edges, e, s, d);
  int rs = b * GENE_NUM + s, rd = b * GENE_NUM + d;
  int c = threadIdx.x;          // 0..127
  int h = c >> 6;
  float v = es[rs * 2 + h] + ed[rd * 2 + h];
  v = v > 0.f ? v : 0.2f * v;
  float ee = __expf(v - funkey(maxk[rd * 2 + h]));
  float alpha = ee / (denom[rd * 2 + h] + 1e-16f);
  atomicAdd(&out[(size_t)rd * HIDDEN + c], alpha * xw[(size_t)rs * HIDDEN + c]);
}

// relu(LN_128(aggr + bias)), one wave per row
__global__ void post_gat_kernel(const float* __restrict__ aggr, const float* __restrict__ bias,
                                const float* __restrict__ g, const float* __restrict__ bb,
                                float* __restrict__ out) {
  int row = blockIdx.x, lane = threadIdx.x;
  float x[4];
  float s = 0.f;
#pragma unroll
  for (int i = 0; i < 4; ++i) {
    x[i] = aggr[(size_t)row * 128 + lane + 32 * i] + bias[lane + 32 * i];
    s += x[i];
  }
  float mean = wave_sum(s) / 128.f;
  float vs = 0.f;
#pragma unroll
  for (int i = 0; i < 4; ++i) { float t = x[i] - mean; vs += t * t; }
  float rs = rsqrtf(wave_sum(vs) / 128.f + EPS);
#pragma unroll
  for (int i = 0; i < 4; ++i) {
    float v = (x[i] - mean) * rs * g[lane + 32 * i] + bb[lane + 32 * i];
    out[(size_t)row * 128 + lane + 32 * i] = v > 0.f ? v : 0.f;
  }
}

// gwf[b,n] = dot(res[b,n,:], indv_w[n,:])
__global__ void final_dot_kernel(const float* __restrict__ res, const float* __restrict__ w,
                                 float* __restrict__ out) {
  int t = blockIdx.x * blockDim.x + threadIdx.x;
  if (t >= NROWS) return;
  int n = t % GENE_NUM;
  float s = 0.f;
  for (int c = 0; c < HIDDEN; ++c) s += res[(size_t)t * HIDDEN + c] * w[(size_t)n * HIDDEN + c];
  out[t] = s;
}

extern "C" void kernel_launch(void* const* d_in, const int* in_sizes, int n_in,
                              void* d_out, int out_size, void* d_ws, size_t ws_size,
                              hipStream_t stream) {
  // setup_inputs() order: x_t, x_noisy, time_emb, cell_type_emb, edge_index, then params
  const float* x_noisy  = (const float*)d_in[1];
  const float* ct_emb   = (const float*)d_in[3];
  const int*   ei       = (const int*)d_in[4];
  const float* gene_emb = (const float*)d_in[5];
  const float* ln_g     = (const float*)d_in[6];
  const float* ln_b     = (const float*)d_in[7];
  const float* indv_w   = (const float*)d_in[8];
  const float *xn_w0 = (const float*)d_in[9],  *xn_b0 = (const float*)d_in[10];
  const float *xn_w1 = (const float*)d_in[11], *xn_b1 = (const float*)d_in[12];
  const float *xn_w2 = (const float*)d_in[13], *xn_b2 = (const float*)d_in[14];
  const float *ct_w0 = (const float*)d_in[15], *ct_b0 = (const float*)d_in[16];
  const float *ct_w1 = (const float*)d_in[17], *ct_b1 = (const float*)d_in[18];
  const float *g0_w = (const float*)d_in[19], *g0_asrc = (const float*)d_in[20],
              *g0_adst = (const float*)d_in[21], *g0_b = (const float*)d_in[22],
              *gn0_g = (const float*)d_in[23], *gn0_b = (const float*)d_in[24];
  const float *g1_w = (const float*)d_in[25], *g1_asrc = (const float*)d_in[26],
              *g1_adst = (const float*)d_in[27], *g1_b = (const float*)d_in[28],
              *gn1_g = (const float*)d_in[29], *gn1_b = (const float*)d_in[30];
  const float *ge_w0 = (const float*)d_in[31], *ge_b0 = (const float*)d_in[32];
  const float *ge_w1 = (const float*)d_in[33], *ge_b1 = (const float*)d_in[34];
  const float *ge_w2 = (const float*)d_in[35], *ge_b2 = (const float*)d_in[36];
  float* out = (float*)d_out;
  (void)n_in; (void)out_size; (void)ws_size;

  int n_edges  = in_sizes[4] / 2;
  int etot     = n_edges + GENE_NUM;
  int etot_all = BATCH * etot;

  float* ws = (float*)d_ws;
  float* buf64a = ws;             ws += NROWS * 64;
  float* buf64b = ws;             ws += NROWS * 64;
  float* lnge   = ws;             ws += GENE_NUM * 64;
  float* ctln   = ws;             ws += 256;
  float* es     = ws;             ws += NROWS * 2;
  float* ed     = ws;             ws += NROWS * 2;
  unsigned* maxk = (unsigned*)ws; ws += NROWS * 2;
  float* denom  = ws;             ws += NROWS * 2;
  float* buf128a = ws;            ws += NROWS * 128;
  float* buf128b = ws;            ws += NROWS * 128;
  float* buf128c = ws;            ws += NROWS * 128;

  dim3 w32(32);

  // cell-type branch (tiny) + x_noisy encoder + gene embedding LN -> gi
  ct_mlp_ln_kernel<<<1, 64, 0, stream>>>(ct_emb, ct_w0, ct_b0, ct_w1, ct_b1, ln_g, ln_b, ctln);
  xn0_kernel<<<(NROWS * 64 + 255) / 256, 256, 0, stream>>>(x_noisy, xn_w0, xn_b0, buf64a);
  wmma_gemm_f32<<<dim3(NROWS / 16, 1), w32, 0, stream>>>(buf64a, xn_w1, xn_b1, buf64b, 64, 64, 1);
  wmma_gemm_f32<<<dim3(NROWS / 16, 1), w32, 0, stream>>>(buf64b, xn_w2, xn_b2, buf64a, 64, 64, 0);
  ln_rows64_kernel<<<GENE_NUM, w32, 0, stream>>>(gene_emb, ln_g, ln_b, lnge);
  combine_gi_kernel<<<NROWS, w32, 0, stream>>>(buf64a, lnge, ctln, ln_g, ln_b, buf64b);

  // two GAT layers
  for (int layer = 0; layer < 2; ++layer) {
    const float* W   = layer ? g1_w : g0_w;
    const float* asr = layer ? g1_asrc : g0_asrc;
    const float* ads = layer ? g1_adst : g0_adst;
    const float* gb  = layer ? g1_b : g0_b;
    const float* ng  = layer ? gn1_g : gn0_g;
    const float* nb  = layer ? gn1_b : gn0_b;
    const float* xin = layer ? buf128c : buf64b;
    int K = layer ? 128 : 64;
    wmma_gemm_f32<<<dim3(NROWS / 16, 2), w32, 0, stream>>>(xin, W, nullptr, buf128a, K, 128, 0);
    attn_coef_kernel<<<(NROWS * 2 + 255) / 256, 256, 0, stream>>>(buf128a, asr, ads, es, ed);
    zero_kernel<<<(NROWS * 2 + 255) / 256, 256, 0, stream>>>((float*)maxk, NROWS * 2);
    zero_kernel<<<(NROWS * 2 + 255) / 256, 256, 0, stream>>>(denom, NROWS * 2);
    zero_kernel<<<(NROWS * 128 + 255) / 256, 256, 0, stream>>>(buf128b, NROWS * 128);
    edge_max_kernel<<<(etot_all + 255) / 256, 256, 0, stream>>>(ei, n_edges, es, ed, maxk);
    edge_denom_kernel<<<(etot_all + 255) / 256, 256, 0, stream>>>(ei, n_edges, es, ed, maxk, denom);
    edge_scatter_kernel<<<etot_all, 128, 0, stream>>>(ei, n_edges, es, ed, maxk, denom, buf128a, buf128b);
    post_gat_kernel<<<NROWS, w32, 0, stream>>>(buf128b, gb, ng, nb, buf128c);
  }

  // gene encoder MLP (silu, silu, none) + final per-gene dot
  wmma_gemm_f32<<<dim3(NROWS / 16, 2), w32, 0, stream>>>(buf128c, ge_w0, ge_b0, buf128a, 128, 128, 1);
  wmma_gemm_f32<<<dim3(NROWS / 16, 2), w32, 0, stream>>>(buf128a, ge_w1, ge_b1, buf128b, 128, 128, 1);
  wmma_gemm_f32<<<dim3(NROWS / 16, 2), w32, 0, stream>>>(buf128b, ge_w2, ge_b2, buf128a, 128, 128, 0);
  final_dot_kernel<<<(NROWS + 255) / 256, 256, 0, stream>>>(buf128a, indv_w, out);
}